// LovaszSoftmaxLoss_45165876085193
// MI455X (gfx1250) — compile-verified
//
#include <hip/hip_runtime.h>
#include <hip/hip_bf16.h>
#include <stdint.h>

#define BSZ   4
#define CSZ   19
#define NPIX  (512*512)          // 2^18
#define NSEG  (BSZ*CSZ)          // 76
#define NBINS 8192
#define ESCALE 16777216.0f       // 2^24 fixed-point scale for error sums

typedef __attribute__((ext_vector_type(2))) float v2f;
typedef __attribute__((ext_vector_type(8))) float v8f;

// ---------------------------------------------------------------------------
// CDNA5 async global->LDS helpers (ASYNCcnt path, CDNA5 ISA)
// Per-lane: 32-bit LDS dest address VGPR, 64-bit global address VGPR pair.
// Generic pointers to __shared__ have LDS offset in addr[31:0] (flat aperture).
// ---------------------------------------------------------------------------
__device__ __forceinline__ void async_ld_b128(void* ldsp, const void* gaddr)
{
    uint32_t lds32 = (uint32_t)(uintptr_t)ldsp;
    asm volatile("global_load_async_to_lds_b128 %0, %1, off"
                 :: "v"(lds32), "v"(gaddr) : "memory");
}
__device__ __forceinline__ void wait_async_le(int keep4)
{
    if (keep4) asm volatile("s_wait_asynccnt 0x4" ::: "memory");
    else       asm volatile("s_wait_asynccnt 0x0" ::: "memory");
}

// ---------------------------------------------------------------------------
// Kernel 1: fused softmax + |p - fg| error histogramming (counting sort).
// Integer/fixed-point atomics only -> bit-deterministic histograms (hit L2;
// the whole 10 MB histogram fits easily in the 192 MB global L2).
// ---------------------------------------------------------------------------
__global__ __launch_bounds__(256) void softmax_hist_kernel(
    const float* __restrict__ logits, const int* __restrict__ target,
    unsigned int* __restrict__ cnt, unsigned int* __restrict__ fgc,
    unsigned long long* __restrict__ esum)
{
    int idx = blockIdx.x * 256 + threadIdx.x;      // [0, B*N)
    int b = idx >> 18;
    int n = idx & (NPIX - 1);
    int lab = target[((size_t)b << 18) + n];

    const float* lp = logits + (((size_t)b * CSZ) << 18) + n;
    float ex[CSZ];
    float mx = -3.4e38f;
#pragma unroll
    for (int c = 0; c < CSZ; ++c) {
        ex[c] = lp[(size_t)c << 18];               // coalesced across lanes
        mx = fmaxf(mx, ex[c]);
    }
    float s = 0.0f;
#pragma unroll
    for (int c = 0; c < CSZ; ++c) { ex[c] = __expf(ex[c] - mx); s += ex[c]; }
    float inv = 1.0f / s;

#pragma unroll
    for (int c = 0; c < CSZ; ++c) {
        float p = ex[c] * inv;
        bool isfg = (lab == c);
        float err = isfg ? (1.0f - p) : p;
        err = fminf(fmaxf(err, 0.0f), 1.0f);
        int bin = (int)(err * (float)NBINS);
        bin = (bin > NBINS - 1) ? (NBINS - 1) : bin;
        size_t base = ((size_t)(b * CSZ + c) * NBINS) + (size_t)bin;
        atomicAdd(&cnt[base], 1u);
        if (isfg) atomicAdd(&fgc[base], 1u);
        atomicAdd(&esum[base], (unsigned long long)(err * ESCALE));
    }
}

// ---------------------------------------------------------------------------
// Tensor-core prefix scan: inclusive column scan of a 16x16 f32 matrix via
// D = L x V with L = lower-triangular ones, chained K=4 WMMAs (K total 16).
// vals/colpre are column-major: element (row r, col c) at [c*16 + r].
// Executed by one full wave (EXEC all ones). Exact for ints < 2^24.
// ---------------------------------------------------------------------------
__device__ inline void wmma_colscan(const float* __restrict__ vals,
                                    float* __restrict__ colpre, int lane)
{
    int nm = lane & 15;                 // A row (M) and B/C/D column (N)
    int kb = (lane < 16) ? 0 : 2;       // K slots held by this lane half
    v8f acc = {0.f,0.f,0.f,0.f,0.f,0.f,0.f,0.f};
#pragma unroll
    for (int j = 0; j < 4; ++j) {
        v2f a, bmat;
        a.x = (nm >= 4*j + kb    ) ? 1.0f : 0.0f;   // L[m][k]
        a.y = (nm >= 4*j + kb + 1) ? 1.0f : 0.0f;
        bmat.x = vals[nm*16 + 4*j + kb    ];        // V[k][n]
        bmat.y = vals[nm*16 + 4*j + kb + 1];
        acc = __builtin_amdgcn_wmma_f32_16x16x4_f32(
            false, a, false, bmat, (short)0, acc, false, false);
    }
    int radd = (lane < 16) ? 0 : 8;     // C/D: VGPR g holds rows g and g+8
#pragma unroll
    for (int g = 0; g < 8; ++g)
        colpre[nm*16 + g + radd] = acc[g];
}

// ---------------------------------------------------------------------------
// Kernel 2: per-(b,c) Lovasz loss from the descending-order histogram.
// One block per segment. Double-buffered async global->LDS staging of the
// histogram chunks (1024 bins = 16 KB) overlapped with the WMMA block scan.
// ---------------------------------------------------------------------------
__global__ __launch_bounds__(256) void lovasz_seg_kernel(
    const unsigned int* __restrict__ cnt, const unsigned int* __restrict__ fgc,
    const unsigned long long* __restrict__ esum, float* __restrict__ seg_loss)
{
    __shared__ __align__(16) unsigned int        cnt_s[2][1024];   //  8 KB
    __shared__ __align__(16) unsigned int        fgc_s[2][1024];   //  8 KB
    __shared__ __align__(16) unsigned long long  es_s [2][1024];   // 16 KB
    __shared__ float vals_m[256], vals_f[256], pre_m[256], pre_f[256];
    __shared__ float red[256];

    const int tid = threadIdx.x;
    const int seg = blockIdx.x;
    const size_t base = (size_t)seg * NBINS;
    const int CHUNK = 1024, PT = 4, NCH = NBINS / CHUNK;   // 8 chunks

    // Issue async staging of chunk 0 (descending order: chunk ch covers global
    // bins [NBINS-(ch+1)*CHUNK, NBINS-ch*CHUNK), stored ascending in LDS).
    {
        size_t g = base + (size_t)(NBINS - CHUNK) + tid * 4;
        async_ld_b128(&cnt_s[0][tid * 4], &cnt[g]);
        async_ld_b128(&fgc_s[0][tid * 4], &fgc[g]);
        async_ld_b128(&es_s[0][tid * 4],     &esum[g]);
        async_ld_b128(&es_s[0][tid * 4 + 2], &esum[g + 2]);
    }

    // G = total foreground count of this segment (overlaps with async load)
    unsigned int gl = 0;
    for (int i = tid; i < NBINS; i += 256) gl += fgc[base + i];
    red[tid] = (float)gl;
    __syncthreads();
    for (int s = 128; s > 0; s >>= 1) { if (tid < s) red[tid] += red[tid + s]; __syncthreads(); }
    float G = red[0];
    __syncthreads();

    const int r = tid & 15, c = tid >> 4; // scan matrix coords (column-major)
    float Rc = 0.f, Fc = 0.f, loss = 0.f; // carried exclusive rank / fg count

    for (int ch = 0; ch < NCH; ++ch) {
        const int buf = ch & 1;
        if (ch + 1 < NCH) {               // prefetch next chunk into other buffer
            int nb = buf ^ 1;
            size_t g = base + (size_t)(NBINS - (ch + 2) * CHUNK) + tid * 4;
            async_ld_b128(&cnt_s[nb][tid * 4], &cnt[g]);
            async_ld_b128(&fgc_s[nb][tid * 4], &fgc[g]);
            async_ld_b128(&es_s[nb][tid * 4],     &esum[g]);
            async_ld_b128(&es_s[nb][tid * 4 + 2], &esum[g + 2]);
        }
        // in-order async completion: <=4 outstanding => current chunk landed
        wait_async_le(ch + 1 < NCH);
        __syncthreads();

        float m_i[PT], f_i[PT];
        unsigned long long e_i[PT];
        float sm = 0.f, sf = 0.f;
#pragma unroll
        for (int i = 0; i < PT; ++i) {    // logical pos -> reversed LDS index
            int li = 1023 - tid * PT - i;
            m_i[i] = (float)cnt_s[buf][li];
            f_i[i] = (float)fgc_s[buf][li];
            e_i[i] = es_s[buf][li];
            sm += m_i[i]; sf += f_i[i];
        }
        vals_m[c*16 + r] = sm;            // V[r][c] = partial of thread c*16+r
        vals_f[c*16 + r] = sf;
        __syncthreads();
        if (tid < 32) {                   // wave 0, EXEC all ones
            wmma_colscan(vals_m, pre_m, tid);
            wmma_colscan(vals_f, pre_f, tid);
        }
        __syncthreads();

        float coff_m = 0.f, coff_f = 0.f, tot_m = 0.f, tot_f = 0.f;
#pragma unroll
        for (int cc = 0; cc < 16; ++cc) { // column totals + cross-column offset
            float tm = pre_m[cc*16 + 15], tf = pre_f[cc*16 + 15];
            if (cc < c) { coff_m += tm; coff_f += tf; }
            tot_m += tm; tot_f += tf;
        }
        float R = Rc + coff_m + ((r > 0) ? pre_m[c*16 + r - 1] : 0.f);
        float F = Fc + coff_f + ((r > 0) ? pre_f[c*16 + r - 1] : 0.f);
#pragma unroll
        for (int i = 0; i < PT; ++i) {
            float m = m_i[i];
            if (m > 0.f) {                // tie-group contribution
                float Rn = R + m, Fn = F + f_i[i];
                float J0 = (R > 0.f) ? (1.f - (G - F ) / (G + R  - F )) : 0.f;
                float J1 = 1.f - (G - Fn) / (G + Rn - Fn);
                float ebar = (float)e_i[i] * (1.0f / ESCALE) / m;
                loss += ebar * (J1 - J0);
                R = Rn; F = Fn;
            }
        }
        Rc += tot_m; Fc += tot_f;
        __syncthreads();                  // buffer reuse fence
    }

    red[tid] = loss;
    __syncthreads();
    for (int s = 128; s > 0; s >>= 1) { if (tid < s) red[tid] += red[tid + s]; __syncthreads(); }
    if (tid == 0) seg_loss[seg] = red[0];
}

// ---------------------------------------------------------------------------
// Kernel 3: deterministic final mean over 76 segments.
// ---------------------------------------------------------------------------
__global__ void final_reduce_kernel(const float* __restrict__ seg_loss,
                                    float* __restrict__ out)
{
    if (blockIdx.x == 0 && threadIdx.x == 0) {
        float s = 0.f;
        for (int i = 0; i < NSEG; ++i) s += seg_loss[i];
        out[0] = s / (float)NSEG;
    }
}

extern "C" void kernel_launch(void* const* d_in, const int* in_sizes, int n_in,
                              void* d_out, int out_size, void* d_ws, size_t ws_size,
                              hipStream_t stream)
{
    const float* logits = (const float*)d_in[0];
    const int*   target = (const int*)d_in[1];

    char* ws = (char*)d_ws;
    const size_t nh = (size_t)NSEG * NBINS;            // 622592 bins
    unsigned long long* esum = (unsigned long long*)ws;            // 8B each
    unsigned int* cnt = (unsigned int*)(ws + nh * 8);              // 4B each
    unsigned int* fgc = (unsigned int*)(ws + nh * 12);             // 4B each
    float* seg_loss   = (float*)(ws + nh * 16);                    // 76 floats
    // total ws use: ~9.97 MB

    hipMemsetAsync(d_ws, 0, nh * 16, stream);          // zero histograms

    softmax_hist_kernel<<<(BSZ * NPIX) / 256, 256, 0, stream>>>(
        logits, target, cnt, fgc, esum);
    lovasz_seg_kernel<<<NSEG, 256, 0, stream>>>(cnt, fgc, esum, seg_loss);
    final_reduce_kernel<<<1, 32, 0, stream>>>(seg_loss, (float*)d_out);
}